// GATLayer_20521353740693
// MI455X (gfx1250) — compile-verified
//
#include <hip/hip_runtime.h>
#include <math.h>

#define NN 8192
#define FD 128
#define SLOPE 0.2f

typedef float v2f __attribute__((ext_vector_type(2)));
typedef float v8f __attribute__((ext_vector_type(8)));

__device__ __forceinline__ float lrelu(float x) { return x > 0.f ? x : SLOPE * x; }

__device__ __forceinline__ v8f wmma4(v2f a, v2f b, v8f c) {
    // V_WMMA_F32_16X16X4_F32: D = A(16x4) * B(4x16) + C(16x16), f32, wave32
    return __builtin_amdgcn_wmma_f32_16x16x4_f32(false, a, false, b, (short)0, c, false, false);
}

// ---------------------------------------------------------------------------
// Tensor Data Mover: DMA a 64x128 f32 tile (rows of inp) from global -> LDS.
// Builds the D# descriptor per CDNA5 ISA:
//   group0: count=1 | lds_addr | global_addr[56:0] | type=2
//   group1: data_size=4B, tensor_dim0=128, tensor_dim1=8192,
//           tile_dim0=128, tile_dim1=64, tensor_dim0_stride=128
//   group2/3: zero (2-D tensor)
// Tracked with TENSORcnt; completed via s_wait_tensorcnt.
// This toolchain exposes the 6-arg builtin:
//   (uint32x4 g0, int32x8 g1, int32x4 g2, int32x4 g3, int32x8 pad, i32 cpol)
// ---------------------------------------------------------------------------
#if __has_builtin(__builtin_amdgcn_tensor_load_to_lds)
#define USE_TDM 1
typedef unsigned int v4u __attribute__((ext_vector_type(4)));
typedef int v8i __attribute__((ext_vector_type(8)));
typedef int v4i __attribute__((ext_vector_type(4)));

__device__ __forceinline__ void tdm_load_tile64x128(const float* gsrc, unsigned int lds_off) {
    const unsigned long long ga = (unsigned long long)(uintptr_t)gsrc;
    v4u g0;
    g0.x = 1u;                                                  // count=1, user D#
    g0.y = lds_off;                                             // lds_addr (bytes)
    g0.z = (unsigned int)ga;                                    // global_addr[31:0]
    g0.w = (unsigned int)((ga >> 32) & 0x01FFFFFFu) | (2u << 30); // addr[56:32] | type=2
    v8i g1;
    g1[0] = (int)(2u << 16);          // wg_mask=0, data_size=2 (4 bytes), no flags
    g1[1] = (int)(128u << 16);        // bits[63:48]  = tensor_dim0 lo = 128
    g1[2] = (int)(8192u << 16);       // bits[95:80]  = tensor_dim1 lo = 8192 (dim0 hi = 0)
    g1[3] = (int)(128u << 16);        // bits[127:112]= tile_dim0 = 128   (dim1 hi = 0)
    g1[4] = 64;                       // bits[143:128]= tile_dim1 = 64, tile_dim2 = 0
    g1[5] = 128;                      // tensor_dim0_stride = 128 (low 32 of 48)
    g1[6] = 0;                        // stride hi | tensor_dim1_stride lo (unused, 2D)
    g1[7] = 0;
    v4i gz4 = {0, 0, 0, 0};
    v8i gz8 = {0, 0, 0, 0, 0, 0, 0, 0};
    __builtin_amdgcn_tensor_load_to_lds(g0, g1, gz4, gz4, gz8, 0);
}
#else
#define USE_TDM 0
#endif

// ---------------------------------------------------------------------------
// Kernel 1: H = inp @ W   (8192x128 @ 128x128, f32 WMMA)
// block = 256 threads = 8 waves; wave handles 16 rows x 128 cols (8 C tiles)
// ---------------------------------------------------------------------------
__global__ __launch_bounds__(256) void k_gemm_h(const float* __restrict__ A,
                                                const float* __restrict__ W,
                                                float* __restrict__ H) {
    const int lane = threadIdx.x & 31;
    const int wave = threadIdx.x >> 5;
    const int m    = lane & 15;
    const int kk   = (lane >> 4) << 1;   // 0 or 2
    const int row  = blockIdx.x * 128 + wave * 16 + m;

    const v8f vz = {0.f, 0.f, 0.f, 0.f, 0.f, 0.f, 0.f, 0.f};
    v8f acc[8];
#pragma unroll
    for (int n = 0; n < 8; ++n) acc[n] = vz;

    const float* ar = A + (size_t)row * FD;
    for (int k0 = 0; k0 < FD; k0 += 4) {
        v2f av = {ar[k0 + kk], ar[k0 + kk + 1]};
        const float* wp = W + (size_t)(k0 + kk) * FD + m;
#pragma unroll
        for (int n = 0; n < 8; ++n) {
            v2f bv = {wp[n * 16], wp[n * 16 + FD]};
            acc[n] = wmma4(av, bv, acc[n]);
        }
    }

    const int rbase = blockIdx.x * 128 + wave * 16 + (lane >> 4) * 8;
#pragma unroll
    for (int n = 0; n < 8; ++n)
#pragma unroll
        for (int r = 0; r < 8; ++r)
            H[(size_t)(rbase + r) * FD + n * 16 + m] = acc[n][r];
}

// ---------------------------------------------------------------------------
// Kernel 2: f1[i] = H[i,:] . a[0:128],  f2[i] = H[i,:] . a[128:256]
// ---------------------------------------------------------------------------
__global__ __launch_bounds__(256) void k_f12(const float* __restrict__ H,
                                             const float* __restrict__ a,
                                             float* __restrict__ f1,
                                             float* __restrict__ f2) {
    const int lane = threadIdx.x & 31;
    const int wave = threadIdx.x >> 5;
    const int row  = blockIdx.x * 8 + wave;
    const float* hr = H + (size_t)row * FD;
    float s1 = 0.f, s2 = 0.f;
#pragma unroll
    for (int t = 0; t < 4; ++t) {
        int c = lane + t * 32;
        float hv = hr[c];
        s1 += hv * a[c];
        s2 += hv * a[FD + c];
    }
#pragma unroll
    for (int off = 16; off; off >>= 1) {
        s1 += __shfl_xor(s1, off, 32);
        s2 += __shfl_xor(s2, off, 32);
    }
    if (lane == 0) { f1[row] = s1; f2[row] = s2; }
}

// ---------------------------------------------------------------------------
// Kernel 3: f2max = max(f2)  (single block reduce)
// ---------------------------------------------------------------------------
__global__ __launch_bounds__(1024) void k_f2max(const float* __restrict__ f2,
                                                float* __restrict__ outv) {
    __shared__ float red[1024];
    float v = -3.4e38f;
    for (int i = threadIdx.x; i < NN; i += 1024) v = fmaxf(v, f2[i]);
    red[threadIdx.x] = v;
    __syncthreads();
    for (int s = 512; s; s >>= 1) {
        if (threadIdx.x < s) red[threadIdx.x] = fmaxf(red[threadIdx.x], red[threadIdx.x + s]);
        __syncthreads();
    }
    if (threadIdx.x == 0) outv[0] = red[0];
}

// ---------------------------------------------------------------------------
// Kernel 4: fused masked-softmax attention + aggregation + alpha-mix
//   p_ij = adj>0 ? exp(lrelu(f1_i+f2_j) - m_i) : 0   (m_i = lrelu(f1_i+max f2))
//   hi   = (P @ inp) / rowsum(P)
//   sup  = (1-alpha)*hi + alpha*h0
// block: 256 thr = 8 waves, 128 rows/block; streams adj once from HBM,
// inp tiles (64x128) DMA'd into LDS by the Tensor Data Mover (wave 0 only)
// and consumed by all 8 waves via WMMA B-fragments.
// ---------------------------------------------------------------------------
__global__ __launch_bounds__(256) void k_att(const float* __restrict__ inp,
                                             const float* __restrict__ adj,
                                             const float* __restrict__ f1,
                                             const float* __restrict__ f2,
                                             const float* __restrict__ f2max,
                                             const float* __restrict__ h0,
                                             const float* __restrict__ alpha_p,
                                             float* __restrict__ sup) {
    __shared__ float sh_inp[64 * FD];   // 32 KB tile of inp rows
    __shared__ float sh_rs[128];        // per-row softmax denominators

    const int lane = threadIdx.x & 31;
    const int wave = threadIdx.x >> 5;
    const int m    = lane & 15;
    const int kk   = (lane >> 4) << 1;
    const int row  = blockIdx.x * 128 + wave * 16 + m;

    const float f1r   = f1[row];
    const float mrow  = lrelu(f1r + f2max[0]);
    const float alpha = alpha_p[0];
    const float* arow = adj + (size_t)row * NN;

#if USE_TDM
    // LDS aperture: flat address low 32 bits == LDS byte offset
    const unsigned int lds_off = (unsigned int)(uintptr_t)(void*)sh_inp;
#endif

    const v8f vz = {0.f, 0.f, 0.f, 0.f, 0.f, 0.f, 0.f, 0.f};
    v8f acc[8];
#pragma unroll
    for (int n = 0; n < 8; ++n) acc[n] = vz;
    float rs = 0.f;

    for (int jb = 0; jb < NN; jb += 64) {
        if (jb + 64 < NN) __builtin_prefetch(arow + jb + 64, 0, 3);
        __syncthreads();   // all waves done reading previous tile
#if USE_TDM
        if (threadIdx.x < 32) {   // one wave issues the tile DMA
            tdm_load_tile64x128(inp + (size_t)jb * FD, lds_off);
            __builtin_amdgcn_s_wait_tensorcnt(0);
        }
#else
        {
            const float4* src = (const float4*)(inp + (size_t)jb * FD);
            float4* dst = (float4*)sh_inp;
#pragma unroll
            for (int t = 0; t < 8; ++t)
                dst[threadIdx.x + t * 256] = src[threadIdx.x + t * 256];
        }
#endif
        __syncthreads();   // tile visible to all waves

        for (int jj = 0; jj < 64; jj += 4) {
            const int j = jb + jj + kk;
            const float2 adjv = *(const float2*)(arow + j);
            float p0 = adjv.x > 0.f ? __expf(lrelu(f1r + f2[j])     - mrow) : 0.f;
            float p1 = adjv.y > 0.f ? __expf(lrelu(f1r + f2[j + 1]) - mrow) : 0.f;
            rs += p0 + p1;
            v2f av = {p0, p1};
            const float* bp = sh_inp + (jj + kk) * FD + m;
#pragma unroll
            for (int n = 0; n < 8; ++n) {
                v2f bv = {bp[n * 16], bp[n * 16 + FD]};
                acc[n] = wmma4(av, bv, acc[n]);
            }
        }
    }

    // fold the two half-wave partial row sums (wave32: xor 16)
    rs += __shfl_xor(rs, 16, 32);
    __syncthreads();
    if (lane < 16) sh_rs[wave * 16 + m] = rs;
    __syncthreads();

    const int rloc = wave * 16 + (lane >> 4) * 8;
#pragma unroll
    for (int r = 0; r < 8; ++r) {
        const float inv  = 1.f / sh_rs[rloc + r];
        const size_t gro = (size_t)(blockIdx.x * 128 + rloc + r) * FD;
#pragma unroll
        for (int n = 0; n < 8; ++n) {
            const int col = n * 16 + m;
            const float hi = acc[n][r] * inv;
            sup[gro + col] = (1.f - alpha) * hi + alpha * h0[gro + col];
        }
    }
}

// ---------------------------------------------------------------------------
// Kernel 5: out = theta*(S @ W) + (1-theta)*S + inp
// ---------------------------------------------------------------------------
__global__ __launch_bounds__(256) void k_gemm_out(const float* __restrict__ S,
                                                  const float* __restrict__ W,
                                                  const float* __restrict__ inp,
                                                  const float* __restrict__ lamda_p,
                                                  const int* __restrict__ l_p,
                                                  float* __restrict__ out) {
    const float theta = fminf(1.f, logf(lamda_p[0] / (float)l_p[0] + 1.f));
    const int lane = threadIdx.x & 31;
    const int wave = threadIdx.x >> 5;
    const int m    = lane & 15;
    const int kk   = (lane >> 4) << 1;
    const int row  = blockIdx.x * 128 + wave * 16 + m;

    const v8f vz = {0.f, 0.f, 0.f, 0.f, 0.f, 0.f, 0.f, 0.f};
    v8f acc[8];
#pragma unroll
    for (int n = 0; n < 8; ++n) acc[n] = vz;

    const float* sr = S + (size_t)row * FD;
    for (int k0 = 0; k0 < FD; k0 += 4) {
        v2f av = {sr[k0 + kk], sr[k0 + kk + 1]};
        const float* wp = W + (size_t)(k0 + kk) * FD + m;
#pragma unroll
        for (int n = 0; n < 8; ++n) {
            v2f bv = {wp[n * 16], wp[n * 16 + FD]};
            acc[n] = wmma4(av, bv, acc[n]);
        }
    }

    const int rbase = blockIdx.x * 128 + wave * 16 + (lane >> 4) * 8;
#pragma unroll
    for (int n = 0; n < 8; ++n)
#pragma unroll
        for (int r = 0; r < 8; ++r) {
            const size_t idx = (size_t)(rbase + r) * FD + n * 16 + m;
            out[idx] = theta * acc[n][r] + (1.f - theta) * S[idx] + inp[idx];
        }
}

// ---------------------------------------------------------------------------
extern "C" void kernel_launch(void* const* d_in, const int* in_sizes, int n_in,
                              void* d_out, int out_size, void* d_ws, size_t ws_size,
                              hipStream_t stream) {
    const float* inp     = (const float*)d_in[0];
    const float* adj     = (const float*)d_in[1];
    const float* h0      = (const float*)d_in[2];
    const float* W       = (const float*)d_in[3];
    const float* a       = (const float*)d_in[4];
    const float* lamda_p = (const float*)d_in[5];
    const float* alpha_p = (const float*)d_in[6];
    const int*   l_p     = (const int*)d_in[7];
    float* out = (float*)d_out;

    float* ws = (float*)d_ws;
    float* H   = ws;                              // N*F
    float* Sup = ws + (size_t)NN * FD;            // N*F
    float* f1  = ws + 2 * (size_t)NN * FD;        // N
    float* f2  = f1 + NN;                         // N
    float* f2m = f2 + NN;                         // 1

    k_gemm_h  <<<NN / 128, 256, 0, stream>>>(inp, W, H);
    k_f12     <<<NN / 8,   256, 0, stream>>>(H, a, f1, f2);
    k_f2max   <<<1,       1024, 0, stream>>>(f2, f2m);
    k_att     <<<NN / 128, 256, 0, stream>>>(inp, adj, f1, f2, f2m, h0, alpha_p, Sup);
    k_gemm_out<<<NN / 128, 256, 0, stream>>>(Sup, W, inp, lamda_p, l_p, out);
}